// NeoGNN_32315333935776
// MI455X (gfx1250) — compile-verified
//
#include <hip/hip_runtime.h>
#include <math.h>

typedef __attribute__((ext_vector_type(2))) float v2f;
typedef __attribute__((ext_vector_type(8))) float v8f;
typedef __attribute__((ext_vector_type(4))) unsigned int v4u;
typedef __attribute__((ext_vector_type(8))) int v8i;
typedef __attribute__((ext_vector_type(4))) int v4i;

#define DEG 16

// ---------------------------------------------------------------------------
// Init: deg = 1.0 (self-loop term), nsf = 0
// ---------------------------------------------------------------------------
__global__ void k_init(float* __restrict__ deg, float* __restrict__ nsf, int N) {
    int i = blockIdx.x * blockDim.x + threadIdx.x;
    if (i < N) { deg[i] = 1.0f; nsf[i] = 0.0f; }
}

// ---------------------------------------------------------------------------
// deg[c] += 1 for every appearance of c in nbr (segment_sum over col)
// ---------------------------------------------------------------------------
__global__ void k_degree(const int* __restrict__ nbr_flat, float* __restrict__ deg, int E) {
    int e = blockIdx.x * blockDim.x + threadIdx.x;
    if (e < E) atomicAdd(&deg[nbr_flat[e]], 1.0f);
}

// ---------------------------------------------------------------------------
// fe scalar MLP (1->8->1) over A_w, scattered into nsf[col]
// ---------------------------------------------------------------------------
__global__ void k_fe_scatter(const float* __restrict__ A_w, const int* __restrict__ nbr_flat,
                             const float* __restrict__ w1, const float* __restrict__ b1,
                             const float* __restrict__ w2, const float* __restrict__ b2,
                             float* __restrict__ nsf, int E) {
    int e = blockIdx.x * blockDim.x + threadIdx.x;
    if (e >= E) return;
    float v = A_w[e];
    float s = b2[0];
#pragma unroll
    for (int j = 0; j < 8; ++j)
        s += w2[j] * fmaxf(fmaf(v, w1[j], b1[j]), 0.0f);
    atomicAdd(&nsf[nbr_flat[e]], s);
}

// ---------------------------------------------------------------------------
// WMMA fp32 GEMM: T[N,C] = X[N,128] @ W[128,C]
// One block per 16-row strip. The 16x128 A-panel is staged into LDS by the
// Tensor Data Mover (wave 0 issues tensor_load_to_lds with a D# descriptor,
// waits TENSORcnt, then the workgroup barrier publishes it). One wave per
// 16-column output tile; K swept in steps of 4 via V_WMMA_F32_16X16X4_F32.
// blockDim.x = 32 * (C/16). Requires N % 16 == 0 (50000 = 3125*16).
// ---------------------------------------------------------------------------
__global__ void k_gemm_wmma(const float* __restrict__ X, const float* __restrict__ W,
                            float* __restrict__ T, int C) {
    __shared__ float As[16 * 128];
    const int m0 = blockIdx.x * 16;
    const int tid = threadIdx.x;

    if (tid < 32) {
        // ---- build Tensor DMA Descriptor (D#) for the 16x128 f32 tile ----
        const unsigned long long ga =
            (unsigned long long)(uintptr_t)(X + (size_t)m0 * 128);
        const unsigned int lds_off = (unsigned int)(uintptr_t)(&As[0]); // addr[31:0] = LDS offset

        v4u g0;
        g0[0] = 1u;                                   // count=1, is_restore=0, gather off
        g0[1] = lds_off;                              // lds_addr (bytes)
        g0[2] = (unsigned int)(ga & 0xFFFFFFFFull);   // global_addr[31:0]
        g0[3] = (unsigned int)((ga >> 32) & 0x01FFFFFFull) | (2u << 30); // addr[56:32], type=2

        v8i g1;
        g1[0] = (int)(2u << 16);          // workgroup_mask=0, data_size=2 (4 bytes)
        g1[1] = (int)(128u << 16);        // tensor_dim0 = 128 (bits 79:48, low half)
        g1[2] = (int)(16u << 16);         // tensor_dim0 hi=0 | tensor_dim1 = 16
        g1[3] = (int)(128u << 16);        // tensor_dim1 hi=0 | tile_dim0 = 128
        g1[4] = (int)16u;                 // tile_dim1 = 16, tile_dim2 = 0
        g1[5] = (int)128u;                // tensor_dim0_stride = 128
        g1[6] = 0;                        // stride hi / dim1_stride lo
        g1[7] = 0;

        v4i gz4 = {0, 0, 0, 0};           // 2D tensor: groups 2/3 unused
        v8i gz8 = {0, 0, 0, 0, 0, 0, 0, 0};

        // clang-23 / amdgpu-toolchain 6-arg form:
        // (uint32x4 g0, int32x8 g1, int32x4, int32x4, int32x8, i32 cpol)
        __builtin_amdgcn_tensor_load_to_lds(g0, g1, gz4, gz4, gz8, 0);
        __builtin_amdgcn_s_wait_tensorcnt(0);
    }
    __syncthreads();

    const int wave = tid >> 5;          // column tile index
    const int lane = tid & 31;
    const int n0   = wave * 16;
    const int lr   = lane & 15;         // M (for A) / N (for B,D)
    const int hi   = lane >> 4;         // half-wave select

    v8f acc = {};
#pragma unroll 4
    for (int k = 0; k < 128; k += 4) {
        v2f a, b;
        // A 16x4: lanes 0-15 hold K=k+0,k+1 ; lanes 16-31 hold K=k+2,k+3
        a.x = As[lr * 128 + k + 2 * hi];
        a.y = As[lr * 128 + k + 2 * hi + 1];
        // B 4x16 (row-major W[k][n]): same half-wave K split, N = lr
        b.x = W[(size_t)(k + 2 * hi)     * C + n0 + lr];
        b.y = W[(size_t)(k + 2 * hi + 1) * C + n0 + lr];
        acc = __builtin_amdgcn_wmma_f32_16x16x4_f32(
            false, a, false, b, (short)0, acc, false, false);
    }

    // D layout: VGPR r -> row m0 + r + 8*hi, col n0 + lr
#pragma unroll
    for (int r = 0; r < 8; ++r)
        T[(size_t)(m0 + r + 8 * hi) * C + n0 + lr] = acc[r];
}

// ---------------------------------------------------------------------------
// GCN aggregation: H[i] = sum_k rsqrt(d_i d_ck) * T[c_k] + T[i]/d_i + bias
// One block per node, one thread per channel -> coalesced neighbor-row loads
// served from L2 (each feature buffer is ~25.6MB << 192MB L2).
// ---------------------------------------------------------------------------
__global__ void k_gcn_agg(const float* __restrict__ T, const int* __restrict__ nbr,
                          const float* __restrict__ deg, const float* __restrict__ bias,
                          float* __restrict__ H, int C, int relu) {
    const int i = blockIdx.x;
    const int f = threadIdx.x;
    const float di   = deg[i];
    const float rsdi = rsqrtf(di);
    float acc = T[(size_t)i * C + f] / di + bias[f];
#pragma unroll
    for (int k = 0; k < DEG; ++k) {
        int c = nbr[(size_t)i * DEG + k];
        float nrm = rsdi * rsqrtf(deg[c]);
        acc = fmaf(nrm, T[(size_t)c * C + f], acc);
    }
    if (relu) acc = fmaxf(acc, 0.0f);
    H[(size_t)i * C + f] = acc;
}

// ---------------------------------------------------------------------------
// out_feat[j] = sum_b H3[e0[b]][j] * H3[e1[b]][j]   (single block, 64 threads)
// ---------------------------------------------------------------------------
__global__ void k_out_feat(const float* __restrict__ H3, const int* __restrict__ edge,
                           float* __restrict__ out_feat, int Bn) {
    const int j = threadIdx.x;   // 0..63
    float acc = 0.0f;
    for (int b = 0; b < Bn; ++b) {
        int e0 = edge[b], e1 = edge[Bn + b];
        acc = fmaf(H3[(size_t)e0 * 64 + j], H3[(size_t)e1 * 64 + j], acc);
    }
    out_feat[j] = acc;
}

// ---------------------------------------------------------------------------
// fn scalar MLP 1->128->1
// ---------------------------------------------------------------------------
__device__ __forceinline__ float mlp128(float x, const float* __restrict__ w1,
                                        const float* __restrict__ b1,
                                        const float* __restrict__ w2, float b2) {
    float s = b2;
#pragma unroll 8
    for (int j = 0; j < 128; ++j)
        s += w2[j] * fmaxf(fmaf(x, w1[j], b1[j]), 0.0f);
    return s;
}

// ---------------------------------------------------------------------------
// Structural path: one thread per batch edge.
// out_struct[b] = sum_j w_dst[j] * sum_i [src_nbr[i]==dst_nbr[j]] w_src[i]
// then gp MLP + sigmoid. Writes sig and raw directly to d_out sections.
// ---------------------------------------------------------------------------
__global__ void k_struct(const int* __restrict__ edge, const int* __restrict__ nbr,
                         const float* __restrict__ A_w, const float* __restrict__ nsf,
                         const float* __restrict__ fn_w1, const float* __restrict__ fn_b1,
                         const float* __restrict__ fn_w2, const float* __restrict__ fn_b2,
                         const float* __restrict__ gp_w1, const float* __restrict__ gp_b1,
                         const float* __restrict__ gp_w2, const float* __restrict__ gp_b2,
                         float* __restrict__ sig_out, float* __restrict__ raw_out, int Bn) {
    int b = blockIdx.x * blockDim.x + threadIdx.x;
    if (b >= Bn) return;
    const int e0 = edge[b];
    const int e1 = edge[Bn + b];

    int   sn[DEG], dn[DEG];
    float ws[DEG], wd[DEG];
#pragma unroll
    for (int k = 0; k < DEG; ++k) {
        sn[k] = nbr[(size_t)e0 * DEG + k];
        dn[k] = nbr[(size_t)e1 * DEG + k];
    }
    const float fb2 = fn_b2[0];
#pragma unroll
    for (int k = 0; k < DEG; ++k) {
        ws[k] = A_w[(size_t)e0 * DEG + k] * mlp128(nsf[sn[k]], fn_w1, fn_b1, fn_w2, fb2);
        wd[k] = A_w[(size_t)e1 * DEG + k] * mlp128(nsf[dn[k]], fn_w1, fn_b1, fn_w2, fb2);
    }
    float os = 0.0f;
#pragma unroll
    for (int j = 0; j < DEG; ++j) {
        float s = 0.0f;
#pragma unroll
        for (int i = 0; i < DEG; ++i)
            if (sn[i] == dn[j]) s += ws[i];
        os = fmaf(s, wd[j], os);
    }
    float raw = mlp128(os, gp_w1, gp_b1, gp_w2, gp_b2[0]);
    raw_out[b] = raw;
    sig_out[b] = 1.0f / (1.0f + expf(-raw));
}

// ---------------------------------------------------------------------------
// Finalize: a = softmax(alpha); out[b][j] = a0*sig[b] + a1*out_feat[j] + 1e-15
// ---------------------------------------------------------------------------
__global__ void k_finalize(const float* __restrict__ sig, const float* __restrict__ ofeat,
                           const float* __restrict__ alpha, float* __restrict__ out, int Bn) {
    float m  = fmaxf(alpha[0], alpha[1]);
    float e0 = expf(alpha[0] - m), e1 = expf(alpha[1] - m);
    float inv = 1.0f / (e0 + e1);
    float a0 = e0 * inv, a1 = e1 * inv;
    int idx = blockIdx.x * blockDim.x + threadIdx.x;
    int total = Bn * 64;
    if (idx < total) {
        int b = idx >> 6, j = idx & 63;
        out[idx] = fmaf(a0, sig[b], fmaf(a1, ofeat[j], 1e-15f));
    }
}

// ---------------------------------------------------------------------------
// Host launcher
// ---------------------------------------------------------------------------
extern "C" void kernel_launch(void* const* d_in, const int* in_sizes, int n_in,
                              void* d_out, int out_size, void* d_ws, size_t ws_size,
                              hipStream_t stream) {
    const int* edge   = (const int*)  d_in[0];
    const float* x    = (const float*)d_in[1];
    const int* nbr    = (const int*)  d_in[2];
    const float* A_w  = (const float*)d_in[3];
    const float* W1   = (const float*)d_in[4];
    const float* b1   = (const float*)d_in[5];
    const float* W2   = (const float*)d_in[6];
    const float* b2   = (const float*)d_in[7];
    const float* W3   = (const float*)d_in[8];
    const float* b3   = (const float*)d_in[9];
    const float* fe_w1 = (const float*)d_in[10];
    const float* fe_b1 = (const float*)d_in[11];
    const float* fe_w2 = (const float*)d_in[12];
    const float* fe_b2 = (const float*)d_in[13];
    const float* fn_w1 = (const float*)d_in[14];
    const float* fn_b1 = (const float*)d_in[15];
    const float* fn_w2 = (const float*)d_in[16];
    const float* fn_b2 = (const float*)d_in[17];
    const float* gp_w1 = (const float*)d_in[18];
    const float* gp_b1 = (const float*)d_in[19];
    const float* gp_w2 = (const float*)d_in[20];
    const float* gp_b2 = (const float*)d_in[21];
    const float* alpha = (const float*)d_in[22];

    const int Bn = in_sizes[0] / 2;       // 1024
    const int N  = in_sizes[2] / DEG;     // 50000
    const int E  = N * DEG;               // 800000

    // workspace layout (floats): deg[N] | nsf[N] | bufA[N*128] | bufB[N*128]
    float* deg  = (float*)d_ws;
    float* nsf  = deg + N;
    float* bufA = nsf + N;
    float* bufB = bufA + (size_t)N * 128;

    float* out      = (float*)d_out;            // [B,64]
    float* sig_out  = out + (size_t)Bn * 64;    // [B]
    float* ofeat    = sig_out + Bn;             // [64]
    float* raw_out  = ofeat + 64;               // [B]

    // 1) degree + nsf init / scatter
    k_init  <<<(N + 255) / 256, 256, 0, stream>>>(deg, nsf, N);
    k_degree<<<(E + 255) / 256, 256, 0, stream>>>(nbr, deg, E);
    k_fe_scatter<<<(E + 255) / 256, 256, 0, stream>>>(A_w, nbr, fe_w1, fe_b1, fe_w2, fe_b2, nsf, E);

    const int strips = N / 16;   // 3125

    // 2) GCN layer 1: x @ W1 -> bufA; aggregate+relu -> bufB
    k_gemm_wmma<<<strips, 32 * 8, 0, stream>>>(x, W1, bufA, 128);
    k_gcn_agg  <<<N, 128, 0, stream>>>(bufA, nbr, deg, b1, bufB, 128, 1);

    // 3) GCN layer 2: bufB @ W2 -> bufA; aggregate+relu -> bufB
    k_gemm_wmma<<<strips, 32 * 8, 0, stream>>>(bufB, W2, bufA, 128);
    k_gcn_agg  <<<N, 128, 0, stream>>>(bufA, nbr, deg, b2, bufB, 128, 1);

    // 4) GCN layer 3 (C=64, no relu): bufB @ W3 -> bufA; aggregate -> bufB
    k_gemm_wmma<<<strips, 32 * 4, 0, stream>>>(bufB, W3, bufA, 64);
    k_gcn_agg  <<<N, 64, 0, stream>>>(bufA, nbr, deg, b3, bufB, 64, 0);

    // 5) out_feat reduction over the batch
    k_out_feat<<<1, 64, 0, stream>>>(bufB, edge, ofeat, Bn);

    // 6) structural path (writes sig/raw straight into d_out sections)
    k_struct<<<(Bn + 127) / 128, 128, 0, stream>>>(edge, nbr, A_w, nsf,
                                                   fn_w1, fn_b1, fn_w2, fn_b2,
                                                   gp_w1, gp_b1, gp_w2, gp_b2,
                                                   sig_out, raw_out, Bn);

    // 7) blend
    k_finalize<<<(Bn * 64 + 255) / 256, 256, 0, stream>>>(sig_out, ofeat, alpha, out, Bn);
}